// FCNNColorCounterValuationFunction_27419071217675
// MI455X (gfx1250) — compile-verified
//
#include <hip/hip_runtime.h>

typedef __attribute__((ext_vector_type(2))) float v2f;
typedef __attribute__((ext_vector_type(8))) float v8f;

// out[i] = a[i, int(z[i, attr])] * 0.999
// Implemented as the reference's one-hot contraction on the f32 WMMA path:
//   D = onehot16x12 * aT12x16  (3 x V_WMMA_F32_16X16X4_F32), take diagonal.
// Bit-exact vs. the gather since one-hot entries are exactly 0/1 in f32.
__global__ __launch_bounds__(256) void FCNNColorCounterValuation_kernel(
    const float* __restrict__ z, const float* __restrict__ a,
    const int* __restrict__ attr_p, float* __restrict__ out, int nrows)
{
    const int lane          = threadIdx.x & 31;          // wave32
    const int wavesPerBlock = blockDim.x >> 5;
    const int wave          = blockIdx.x * wavesPerBlock + (threadIdx.x >> 5);
    const int nwaves        = gridDim.x * wavesPerBlock;
    const int attr          = attr_p[0];

    const int half   = lane >> 4;    // 0: lanes 0-15, 1: lanes 16-31
    const int m      = lane & 15;    // tile row (A) / tile col (B) for this lane
    const int ntiles = nrows >> 4;   // 16 rows per wave-tile

    for (int tile = wave; tile < ntiles; tile += nwaves) {
        const int rowBase = tile << 4;
        const int row     = rowBase + m;

        // idx for row m (both half-waves read the same 16 addresses; dedups in L0)
        const int idx = (int)z[(size_t)row * 16 + attr];
        const float* arow = a + (size_t)row * 10;

        v8f c = {};
#pragma unroll
        for (int kb = 0; kb < 12; kb += 4) {
            const int k0 = kb + (half << 1);
            // A tile (16x4): lane holds onehot entries for K = k0, k0+1
            v2f A;
            A.x = (idx == k0)     ? 1.0f : 0.0f;
            A.y = (idx == k0 + 1) ? 1.0f : 0.0f;
            // B tile (4x16): lane holds a[row][k0..k0+1]; K >= 10 is zero pad.
            // k0 is even and rows have 10 cols, so k0 and k0+1 share validity.
            const bool inb = (k0 < 10);
            v2f Bv = *(const v2f*)(arow + (inb ? k0 : 0));   // always in-bounds
            if (!inb) { Bv.x = 0.0f; Bv.y = 0.0f; }
            c = __builtin_amdgcn_wmma_f32_16x16x4_f32(
                    /*neg_a=*/false, A, /*neg_b=*/false, Bv,
                    /*c_mod=*/(short)0, c, /*reuse_a=*/false, /*reuse_b=*/false);
        }

        // Diagonal D[i][i]: VGPR i&7, lane i (i<8) or lane i+16 (i>=8).
        const int sel = lane & 7;
        float d = c[0];
        d = (sel == 1) ? c[1] : d;
        d = (sel == 2) ? c[2] : d;
        d = (sel == 3) ? c[3] : d;
        d = (sel == 4) ? c[4] : d;
        d = (sel == 5) ? c[5] : d;
        d = (sel == 6) ? c[6] : d;
        d = (sel == 7) ? c[7] : d;

        const bool lo = (lane < 8);
        const bool hi = (lane >= 24);
        if (lo | hi) {
            const int i = lo ? lane : (lane - 16);
            // NT store: don't let the 8MB output evict the ~L2-resident inputs
            __builtin_nontemporal_store(d * 0.999f, out + rowBase + i);
        }
    }

    // Tail rows (nrows % 16): plain gather handled by wave 0.
    if (wave == 0) {
        for (int r = (ntiles << 4) + lane; r < nrows; r += 32) {
            const int idx = (int)z[(size_t)r * 16 + attr];
            __builtin_nontemporal_store(a[(size_t)r * 10 + idx] * 0.999f, out + r);
        }
    }
}

extern "C" void kernel_launch(void* const* d_in, const int* in_sizes, int n_in,
                              void* d_out, int out_size, void* d_ws, size_t ws_size,
                              hipStream_t stream)
{
    const float* z    = (const float*)d_in[0];   // [B, 16] f32
    const float* a    = (const float*)d_in[1];   // [B, 10] f32
    const int*   attr = (const int*)d_in[2];     // scalar attr_index
    float*       out  = (float*)d_out;           // [B] f32

    const int nrows  = in_sizes[1] / 10;         // B
    const int ntiles = nrows / 16;

    const int threads       = 256;               // 8 waves per block
    const int wavesPerBlock = threads / 32;
    int blocks = (ntiles + wavesPerBlock - 1) / wavesPerBlock;
    if (blocks > 8192) blocks = 8192;            // grid-stride over tiles
    if (blocks < 1)    blocks = 1;

    FCNNColorCounterValuation_kernel<<<blocks, threads, 0, stream>>>(
        z, a, attr, out, nrows);
}